// MRD_GNN_47931835023661
// MI455X (gfx1250) — compile-verified
//
#include <hip/hip_runtime.h>
#include <hip/hip_bf16.h>
#include <math.h>

#define N_NODE 100000
#define N_EDGE 625000
#define DIM 128
#define ADIM 64
#define NRELROWS 401
#define BQ 100
#define LDSP 132  // LDS row stride (floats): conflict-free b64 B-operand reads

typedef float v2f __attribute__((ext_vector_type(2)));
typedef float v8f __attribute__((ext_vector_type(8)));

__device__ __forceinline__ v8f wmma4(v2f a, v2f b, v8f c) {
  // D = A(16x4 f32) * B(4x16 f32) + C(16x16 f32)
  return __builtin_amdgcn_wmma_f32_16x16x4_f32(false, a, false, b, (short)0, c,
                                               false, false);
}

__device__ __forceinline__ float sigmoidf_(float x) {
  return 1.0f / (1.0f + __expf(-x));
}

// ---------------------------------------------------------------------------
// Tiny precompute: relWr[r] = rela_embed[r] @ Wr
//                  qWqr[b]  = rela_embed[q_rel[b]] @ Wqr + b_qr
// ---------------------------------------------------------------------------
__global__ void k_small(const float* __restrict__ rela,
                        const float* __restrict__ Wr,
                        const float* __restrict__ Wqr,
                        const float* __restrict__ b_qr,
                        const int* __restrict__ q_rel,
                        float* __restrict__ relWr,
                        float* __restrict__ qWqr) {
  const int t = threadIdx.x;  // 0..63
  if ((int)blockIdx.x < NRELROWS) {
    const int row = blockIdx.x;
    float s = 0.0f;
    for (int k = 0; k < DIM; ++k) s += rela[row * DIM + k] * Wr[k * ADIM + t];
    relWr[row * ADIM + t] = s;
  } else {
    const int bq = blockIdx.x - NRELROWS;
    const int row = q_rel[bq];
    float s = b_qr[t];
    for (int k = 0; k < DIM; ++k) s += rela[row * DIM + k] * Wqr[k * ADIM + t];
    qWqr[bq * ADIM + t] = s;
  }
}

// ---------------------------------------------------------------------------
// hiddenWs = hidden @ Ws   [N,128]x[128,64]
// One wave per 16-row band; Ws staged transposed in LDS once per block.
// ---------------------------------------------------------------------------
__global__ void k_hidden_ws(const float* __restrict__ hidden,
                            const float* __restrict__ Ws,
                            float* __restrict__ out) {
  __shared__ float ldsB[ADIM * LDSP];  // ldsB[n*LDSP + k] = Ws[k][n]  (33.8 KB)
  for (int idx = threadIdx.x; idx < DIM * ADIM; idx += 256) {
    const int k = idx >> 6, n = idx & 63;
    ldsB[n * LDSP + k] = Ws[idx];
  }
  __syncthreads();

  const int wave = (blockIdx.x * blockDim.x + threadIdx.x) >> 5;
  const int m0 = wave * 16;
  if (m0 >= N_NODE) return;  // wave-uniform; no barriers after this point
  const int lane = threadIdx.x & 31;
  const int lrow = lane & 15;
  const int khalf = (lane >> 4) << 1;  // 0 or 2
  int mload = m0 + lrow;
  if (mload >= N_NODE) mload = N_NODE - 1;
  const float* arow = hidden + (size_t)mload * DIM;

  v8f acc0 = {}, acc1 = {}, acc2 = {}, acc3 = {};
#pragma unroll
  for (int ks = 0; ks < 32; ++ks) {
    const int kk = 4 * ks + khalf;
    v2f a = *(const v2f*)(arow + kk);
    v2f b0 = *(const v2f*)&ldsB[(0 * 16 + lrow) * LDSP + kk];
    v2f b1 = *(const v2f*)&ldsB[(1 * 16 + lrow) * LDSP + kk];
    v2f b2 = *(const v2f*)&ldsB[(2 * 16 + lrow) * LDSP + kk];
    v2f b3 = *(const v2f*)&ldsB[(3 * 16 + lrow) * LDSP + kk];
    acc0 = wmma4(a, b0, acc0);
    acc1 = wmma4(a, b1, acc1);
    acc2 = wmma4(a, b2, acc2);
    acc3 = wmma4(a, b3, acc3);
  }
  const int rbase = m0 + ((lane >> 4) << 3);
#pragma unroll
  for (int v = 0; v < 8; ++v) {
    const int row = rbase + v;
    if (row < N_NODE) {
      out[(size_t)row * ADIM + 0 + lrow] = acc0[v];
      out[(size_t)row * ADIM + 16 + lrow] = acc1[v];
      out[(size_t)row * ADIM + 32 + lrow] = acc2[v];
      out[(size_t)row * ADIM + 48 + lrow] = acc3[v];
    }
  }
}

// ---------------------------------------------------------------------------
// Per-edge attention + message scatter-add. One wave32 per edge.
// ---------------------------------------------------------------------------
__global__ void k_edge(const float* __restrict__ hidden,
                       const float* __restrict__ rela,
                       const float* __restrict__ hWs,
                       const float* __restrict__ relWr,
                       const float* __restrict__ qWqr,
                       const float* __restrict__ w_alpha,
                       const float* __restrict__ b_alpha,
                       const int* __restrict__ r_idx,
                       const int* __restrict__ rel,
                       const int* __restrict__ sub,
                       const int* __restrict__ obj,
                       float* __restrict__ agg) {
  const int e = (blockIdx.x * blockDim.x + threadIdx.x) >> 5;
  if (e >= N_EDGE) return;
  const int lane = threadIdx.x & 31;
  const int s = sub[e];
  const int r = rel[e];
  const int b = r_idx[e];
  const int o = obj[e];

  float dot = 0.0f;
#pragma unroll
  for (int j = 0; j < 2; ++j) {
    const int a = lane + 32 * j;
    float t = hWs[(size_t)s * ADIM + a] + relWr[r * ADIM + a] + qWqr[b * ADIM + a];
    t = t > 0.0f ? t : 0.0f;  // relu
    dot += t * w_alpha[a];
  }
#pragma unroll
  for (int off = 16; off > 0; off >>= 1) dot += __shfl_xor(dot, off, 32);
  const float alpha = sigmoidf_(dot + b_alpha[0]);

#pragma unroll
  for (int j = 0; j < 4; ++j) {
    const int d = lane + 32 * j;
    const float m = alpha * (hidden[(size_t)s * DIM + d] + rela[(size_t)r * DIM + d]);
    __hip_atomic_fetch_add(&agg[(size_t)o * DIM + d], m, __ATOMIC_RELAXED,
                           __HIP_MEMORY_SCOPE_AGENT);
  }
}

// ---------------------------------------------------------------------------
// hidden_new = relu(agg @ W_h)   [N,128]x[128,128]
// W_h staged in LDS in two 64-column halves (33.8 KB buffer).
// ---------------------------------------------------------------------------
__global__ void k_wh(const float* __restrict__ agg,
                     const float* __restrict__ W_h,
                     float* __restrict__ out) {
  __shared__ float ldsB[ADIM * LDSP];  // ldsB[n*LDSP + k] = W_h[k][half*64 + n]
  const int wave = (blockIdx.x * blockDim.x + threadIdx.x) >> 5;
  const int m0 = wave * 16;
  const int lane = threadIdx.x & 31;
  const int lrow = lane & 15;
  const int khalf = (lane >> 4) << 1;
  int mload = m0 + lrow;
  if (mload >= N_NODE) mload = N_NODE - 1;
  const float* arow = agg + (size_t)mload * DIM;

  v8f acc[8] = {};
  for (int half = 0; half < 2; ++half) {
    // stage 64 columns of W_h, transposed
    for (int idx = threadIdx.x; idx < DIM * ADIM; idx += 256) {
      const int k = idx >> 6, n = idx & 63;
      ldsB[n * LDSP + k] = W_h[k * DIM + half * ADIM + n];
    }
    __syncthreads();
#pragma unroll
    for (int ks = 0; ks < 32; ++ks) {
      const int kk = 4 * ks + khalf;
      v2f a = *(const v2f*)(arow + kk);
#pragma unroll
      for (int t = 0; t < 4; ++t) {
        v2f b = *(const v2f*)&ldsB[(t * 16 + lrow) * LDSP + kk];
        acc[half * 4 + t] = wmma4(a, b, acc[half * 4 + t]);
      }
    }
    __syncthreads();
  }

  const int rbase = m0 + ((lane >> 4) << 3);
#pragma unroll
  for (int t = 0; t < 8; ++t) {
#pragma unroll
    for (int v = 0; v < 8; ++v) {
      const int row = rbase + v;
      if (row < N_NODE) {
        const float x = acc[t][v];
        out[(size_t)row * DIM + t * 16 + lrow] = x > 0.0f ? x : 0.0f;
      }
    }
  }
}

// ---------------------------------------------------------------------------
// Fused GRU: gi = hn@Wih.T + bih ; gh = h0@Whh.T + bhh ; gates ; h_out
// One wave per 16-row band; A operands held in registers across col tiles;
// per-tile gate weights (6 slabs of 16 cols) staged in LDS per block.
// ---------------------------------------------------------------------------
__global__ void k_gru(const float* __restrict__ hn, const float* __restrict__ h0,
                      const float* __restrict__ Wih, const float* __restrict__ Whh,
                      const float* __restrict__ bih, const float* __restrict__ bhh,
                      float* __restrict__ out) {
  __shared__ float ldsB[6 * 16 * LDSP];  // 50.7 KB: slabs {ih_r,ih_z,ih_n,hh_r,hh_z,hh_n}
  const int wave = (blockIdx.x * blockDim.x + threadIdx.x) >> 5;
  const int m0 = wave * 16;
  const int lane = threadIdx.x & 31;
  const int lrow = lane & 15;
  const int khalf = (lane >> 4) << 1;
  int mload = m0 + lrow;
  if (mload >= N_NODE) mload = N_NODE - 1;

  v2f a_hn[32], a_h0[32];
#pragma unroll
  for (int ks = 0; ks < 32; ++ks) {
    const int kk = 4 * ks + khalf;
    a_hn[ks] = *(const v2f*)(hn + (size_t)mload * DIM + kk);
    a_h0[ks] = *(const v2f*)(h0 + (size_t)mload * DIM + kk);
  }
  const int rbase = m0 + ((lane >> 4) << 3);

  for (int t = 0; t < 8; ++t) {
    // stage the 6 weight slabs for this 16-column tile (contiguous row reads)
    for (int idx = threadIdx.x; idx < 6 * 16 * DIM; idx += 256) {
      const int slab = idx >> 11;          // 0..5
      const int r = (idx >> 7) & 15;       // 0..15
      const int k = idx & 127;             // 0..127
      const float* src = (slab < 3) ? Wih : Whh;
      const int gate = (slab < 3) ? slab : slab - 3;
      ldsB[(slab * 16 + r) * LDSP + k] =
          src[(size_t)(gate * DIM + t * 16 + r) * DIM + k];
    }
    __syncthreads();

    const int col = t * 16 + lrow;  // 0..127
    v8f gir = {}, giz = {}, gin = {}, ghr = {}, ghz = {}, ghn = {};
#pragma unroll
    for (int ks = 0; ks < 32; ++ks) {
      const int kk = 4 * ks + khalf;
      v2f br = *(const v2f*)&ldsB[(0 * 16 + lrow) * LDSP + kk];
      v2f bz = *(const v2f*)&ldsB[(1 * 16 + lrow) * LDSP + kk];
      v2f bn = *(const v2f*)&ldsB[(2 * 16 + lrow) * LDSP + kk];
      v2f cr = *(const v2f*)&ldsB[(3 * 16 + lrow) * LDSP + kk];
      v2f cz = *(const v2f*)&ldsB[(4 * 16 + lrow) * LDSP + kk];
      v2f cn = *(const v2f*)&ldsB[(5 * 16 + lrow) * LDSP + kk];
      gir = wmma4(a_hn[ks], br, gir);
      giz = wmma4(a_hn[ks], bz, giz);
      gin = wmma4(a_hn[ks], bn, gin);
      ghr = wmma4(a_h0[ks], cr, ghr);
      ghz = wmma4(a_h0[ks], cz, ghz);
      ghn = wmma4(a_h0[ks], cn, ghn);
    }

    const float bir = bih[col], biz = bih[col + DIM], bin_ = bih[col + 2 * DIM];
    const float bhr = bhh[col], bhz = bhh[col + DIM], bhn_ = bhh[col + 2 * DIM];
#pragma unroll
    for (int v = 0; v < 8; ++v) {
      const int row = rbase + v;
      const int rl = row < N_NODE ? row : N_NODE - 1;
      const float h0v = h0[(size_t)rl * DIM + col];
      const float rg = sigmoidf_(gir[v] + bir + ghr[v] + bhr);
      const float zg = sigmoidf_(giz[v] + biz + ghz[v] + bhz);
      const float ng = tanhf(gin[v] + bin_ + rg * (ghn[v] + bhn_));
      const float ho = (1.0f - zg) * ng + zg * h0v;
      if (row < N_NODE) out[(size_t)row * DIM + col] = ho;
    }
    __syncthreads();  // protect LDS against next tile's staging
  }
}

// ---------------------------------------------------------------------------
extern "C" void kernel_launch(void* const* d_in, const int* in_sizes, int n_in,
                              void* d_out, int out_size, void* d_ws, size_t ws_size,
                              hipStream_t stream) {
  const float* hidden = (const float*)d_in[0];
  const float* rela = (const float*)d_in[1];
  const float* Ws = (const float*)d_in[2];
  const float* Wr = (const float*)d_in[3];
  const float* Wqr = (const float*)d_in[4];
  const float* b_qr = (const float*)d_in[5];
  const float* w_alpha = (const float*)d_in[6];
  const float* b_alpha = (const float*)d_in[7];
  const float* W_h = (const float*)d_in[8];
  const float* W_ih = (const float*)d_in[9];
  const float* W_hh = (const float*)d_in[10];
  const float* b_ih = (const float*)d_in[11];
  const float* b_hh = (const float*)d_in[12];
  const float* h0 = (const float*)d_in[13];
  const int* q_rel = (const int*)d_in[14];
  const int* r_idx = (const int*)d_in[15];
  const int* rel = (const int*)d_in[16];
  const int* sub = (const int*)d_in[17];
  const int* obj = (const int*)d_in[18];
  float* out = (float*)d_out;
  (void)in_sizes; (void)n_in; (void)out_size; (void)ws_size;

  // Workspace partitions (all offsets multiples of 256 bytes)
  float* hWs = (float*)d_ws;                       // N*64
  float* relWr = hWs + (size_t)N_NODE * ADIM;      // 401*64
  float* qWqr = relWr + (size_t)NRELROWS * ADIM;   // 100*64
  float* agg = qWqr + (size_t)((BQ * ADIM + 63) & ~63);  // N*128
  float* hnew = agg + (size_t)N_NODE * DIM;        // N*128

  hipMemsetAsync(agg, 0, (size_t)N_NODE * DIM * sizeof(float), stream);

  k_small<<<NRELROWS + BQ, 64, 0, stream>>>(rela, Wr, Wqr, b_qr, q_rel, relWr, qWqr);

  const int bands = (N_NODE + 15) / 16;   // 6250 waves
  const int blocks = (bands + 7) / 8;     // 8 waves (256 thr) per block

  k_hidden_ws<<<blocks, 256, 0, stream>>>(hidden, Ws, hWs);

  k_edge<<<(int)(((long)N_EDGE * 32 + 255) / 256), 256, 0, stream>>>(
      hidden, rela, hWs, relWr, qWqr, w_alpha, b_alpha, r_idx, rel, sub, obj, agg);

  k_wh<<<blocks, 256, 0, stream>>>(agg, W_h, hnew);

  k_gru<<<blocks, 256, 0, stream>>>(hnew, h0, W_ih, W_hh, b_ih, b_hh, out);
}